// GraphSAINTModel_55662776156290
// MI455X (gfx1250) — compile-verified
//
#include <hip/hip_runtime.h>

typedef __attribute__((ext_vector_type(2))) float v2f;
typedef __attribute__((ext_vector_type(8))) float v8f;

#define N_NODES 50000
#define D_HID   64

// ---------------------------------------------------------------------------
// Degree / normalization
// ---------------------------------------------------------------------------
__global__ void deg_init_kernel(float* __restrict__ deg, int n) {
    int i = blockIdx.x * blockDim.x + threadIdx.x;
    if (i < n) deg[i] = 1.0f;  // self-loop contributes 1
}

__global__ void deg_count_kernel(const long long* __restrict__ dst,
                                 float* __restrict__ deg, int nedges) {
    int e = blockIdx.x * blockDim.x + threadIdx.x;
    if (e < nedges) {
        int d = (int)dst[e];
        unsafeAtomicAdd(&deg[d], 1.0f);
    }
}

__global__ void deg_rsqrt_kernel(float* __restrict__ deg, int n) {
    int i = blockIdx.x * blockDim.x + threadIdx.x;
    if (i < n) deg[i] = __frsqrt_rn(deg[i]);  // deg >= 1 always (self loop)
}

// ---------------------------------------------------------------------------
// WMMA fp32 GEMM: XW[M,64] = act(X[M,K]) @ W[K,64]
// One wave computes a 16x16 C tile via V_WMMA_F32_16X16X4_F32.
// Block = 256 threads = 8 waves -> 32 rows x 64 cols per block.
// ---------------------------------------------------------------------------
template <bool RELU_IN>
__global__ __launch_bounds__(256)
void gemm_wmma_kernel(const float* __restrict__ X, const float* __restrict__ W,
                      float* __restrict__ XW, int M, int K) {
    const int lane  = threadIdx.x & 31;
    const int wave  = threadIdx.x >> 5;     // 0..7
    const int m0    = blockIdx.x * 32 + (wave >> 2) * 16;
    const int n0    = (wave & 3) * 16;

    const int lhalf = lane >> 4;            // 0 or 1: selects K pair
    const int lmod  = lane & 15;            // row (A) / col (B)

    int arow = m0 + lmod;
    if (arow >= M) arow = 0;                // clamp: EXEC must stay all-ones
    const float* __restrict__ Xr = X + (long)arow * K;

    v8f c = {0.f, 0.f, 0.f, 0.f, 0.f, 0.f, 0.f, 0.f};

    for (int k = 0; k < K; k += 4) {
        const int ka = k + 2 * lhalf;
        // A tile 16x4: lane holds X[arow, ka], X[arow, ka+1]
        v2f a;
        a.x = Xr[ka];
        a.y = Xr[ka + 1];
        if (RELU_IN) {
            a.x = fmaxf(a.x, 0.0f);
            a.y = fmaxf(a.y, 0.0f);
        }
        // B tile 4x16: lane holds W[ka, n0+lmod], W[ka+1, n0+lmod]
        v2f b;
        b.x = W[(long)ka * D_HID + n0 + lmod];
        b.y = W[(long)(ka + 1) * D_HID + n0 + lmod];

        c = __builtin_amdgcn_wmma_f32_16x16x4_f32(
            /*neg_a=*/false, a, /*neg_b=*/false, b,
            /*c_mod=*/(short)0, c, /*reuse_a=*/false, /*reuse_b=*/false);
    }

    // C 16x16 f32 layout: VGPR v -> row (m0 + v + 8*lhalf), col (n0 + lmod)
    const int col = n0 + lmod;
#pragma unroll
    for (int v = 0; v < 8; ++v) {
        const int r = m0 + v + 8 * lhalf;
        if (r < M) XW[(long)r * D_HID + col] = c[v];
    }
}

// ---------------------------------------------------------------------------
// out[i,f] = dis[i]^2 * xw[i,f] + b[f]   (self-loop message + bias)
// ---------------------------------------------------------------------------
__global__ void selfloop_bias_kernel(const float* __restrict__ xw,
                                     const float* __restrict__ dis,
                                     const float* __restrict__ bias,
                                     float* __restrict__ out, int n) {
    int t = blockIdx.x * blockDim.x + threadIdx.x;
    int i = t >> 6;          // node
    int f = t & 63;          // feature
    if (i < n) {
        float d = dis[i];
        out[t] = d * d * xw[t] + bias[f];
    }
}

// ---------------------------------------------------------------------------
// Edge aggregation: one wave per edge (grid-stride).
// out[dst] += dis[src]*dis[dst] * xw[src]   over 64 features (2 per lane)
// ---------------------------------------------------------------------------
__global__ __launch_bounds__(256)
void edge_agg_kernel(const long long* __restrict__ src,
                     const long long* __restrict__ dst,
                     const float* __restrict__ dis,
                     const float* __restrict__ xw,
                     float* __restrict__ out, int nedges) {
    const int lane   = threadIdx.x & 31;
    const int gwave  = (blockIdx.x * blockDim.x + threadIdx.x) >> 5;
    const int nwaves = (gridDim.x * blockDim.x) >> 5;

    for (int e = gwave; e < nedges; e += nwaves) {
        const int s = (int)src[e];
        const int d = (int)dst[e];
        const float w = dis[s] * dis[d];
        const float2 v = *(const float2*)(xw + (long)s * D_HID + 2 * lane);
        float* __restrict__ o = out + (long)d * D_HID + 2 * lane;
        unsafeAtomicAdd(o,     w * v.x);
        unsafeAtomicAdd(o + 1, w * v.y);
    }
}

// ---------------------------------------------------------------------------
// Launch
// ---------------------------------------------------------------------------
extern "C" void kernel_launch(void* const* d_in, const int* in_sizes, int n_in,
                              void* d_out, int out_size, void* d_ws, size_t ws_size,
                              hipStream_t stream) {
    const float*     x    = (const float*)d_in[0];            // [N, 128]
    const long long* eidx = (const long long*)d_in[1];        // [2, E] int64
    const float*     W1   = (const float*)d_in[2];
    const float*     b1   = (const float*)d_in[3];
    const float*     W2   = (const float*)d_in[4];
    const float*     b2   = (const float*)d_in[5];
    const float*     W3   = (const float*)d_in[6];
    const float*     b3   = (const float*)d_in[7];

    const int N = in_sizes[0] / 128;       // 50000
    const int E = in_sizes[1] / 2;         // 800000
    const long long* src = eidx;
    const long long* dst = eidx + E;

    // Workspace layout (floats): dis[N] | xw[N*64] | h1[N*64] | h2[N*64]
    float* ws  = (float*)d_ws;
    float* dis = ws;
    float* xw  = dis + N;                  // N*4 bytes is 16B-aligned (200000)
    float* h1  = xw + (size_t)N * D_HID;
    float* h2  = h1 + (size_t)N * D_HID;

    const int TB = 256;
    const int gridN   = (N + TB - 1) / TB;
    const int gridE   = (E + TB - 1) / TB;
    const int gridNF  = (N * D_HID + TB - 1) / TB;
    const int gridGemm = (N + 31) / 32;
    const int gridAgg = 2048;              // grid-stride, 16384 waves

    // --- normalization coefficients ---
    deg_init_kernel<<<gridN, TB, 0, stream>>>(dis, N);
    deg_count_kernel<<<gridE, TB, 0, stream>>>(dst, dis, E);
    deg_rsqrt_kernel<<<gridN, TB, 0, stream>>>(dis, N);

    // --- layer 1: x(128) -> h1(64) ---
    gemm_wmma_kernel<false><<<gridGemm, TB, 0, stream>>>(x, W1, xw, N, 128);
    selfloop_bias_kernel<<<gridNF, TB, 0, stream>>>(xw, dis, b1, h1, N);
    edge_agg_kernel<<<gridAgg, TB, 0, stream>>>(src, dst, dis, xw, h1, E);

    // --- layer 2: relu(h1)(64) -> h2(64) ---
    gemm_wmma_kernel<true><<<gridGemm, TB, 0, stream>>>(h1, W2, xw, N, 64);
    selfloop_bias_kernel<<<gridNF, TB, 0, stream>>>(xw, dis, b2, h2, N);
    edge_agg_kernel<<<gridAgg, TB, 0, stream>>>(src, dst, dis, xw, h2, E);

    // --- layer 3: relu(h2)(64) -> out(64), no activation ---
    float* out = (float*)d_out;
    gemm_wmma_kernel<true><<<gridGemm, TB, 0, stream>>>(h2, W3, xw, N, 64);
    selfloop_bias_kernel<<<gridNF, TB, 0, stream>>>(xw, dis, b3, out, N);
    edge_agg_kernel<<<gridAgg, TB, 0, stream>>>(src, dst, dis, xw, out, E);
}